// ConVIRT_Loss_68521908240536
// MI455X (gfx1250) — compile-verified
//
#include <hip/hip_runtime.h>

typedef float v2f __attribute__((ext_vector_type(2)));
typedef float v8f __attribute__((ext_vector_type(8)));
typedef unsigned int u32;
typedef u32 u32x4 __attribute__((ext_vector_type(4)));
typedef int  i32x4 __attribute__((ext_vector_type(4)));
typedef int  i32x8 __attribute__((ext_vector_type(8)));

#define BATCH 8192
#define DIM   512
#define ALPHA 0.75f
#define INV_TEMP 10.0f
#define EPS 1e-8f

#define BM 128
#define BN 128
#define KB 32
#define LDT 36            // padded LDS row stride in floats (32 data + 4 pad)
#define TILE_F (BM * LDT) // 4608 floats = 18432 bytes per tile buffer

// ---------------------------------------------------------------------------
// TDM: issue a 2D tensor tile DMA (tile_d0=KB floats contiguous, tile_d1=BM
// rows, tensor row stride DIM) from global into LDS at byte offset lds_off.
// D# pad feature inserts 4 DWORDs of padding every 32 DWORDs -> LDT stride.
// ---------------------------------------------------------------------------
__device__ __forceinline__ void tdm_load_tile(unsigned lds_off, const float* gptr) {
    unsigned long long ga = (unsigned long long)(size_t)gptr;
    u32x4 g0;
    g0[0] = 0x1u;                                        // count=1, user mode
    g0[1] = lds_off;                                     // lds_addr (bytes)
    g0[2] = (u32)(ga & 0xffffffffu);                     // global_addr[31:0]
    g0[3] = (u32)((ga >> 32) & 0x01ffffffu) | (2u << 30);// addr[56:32] | type=2
    i32x8 g1;
    g1[0] = (2 << 16)        // data_size = 4 bytes
          | (1 << 20)        // pad_enable
          | (4 << 22)        // pad_interval: 32 DWORDs
          | (3 << 25);       // pad_amount:   4 DWORDs
    g1[1] = (DIM & 0xffff) << 16;                        // tensor_dim0 lo16
    g1[2] = ((DIM >> 16) & 0xffff) | ((BATCH & 0xffff) << 16); // dim0 hi | dim1 lo
    g1[3] = ((BATCH >> 16) & 0xffff) | (KB << 16);       // dim1 hi | tile_dim0
    g1[4] = BM;                                          // tile_dim1 (tile_dim2=0)
    g1[5] = DIM;                                         // tensor_dim0_stride lo32
    g1[6] = 0;
    g1[7] = 0;
    i32x4 z4 = {0, 0, 0, 0};                             // groups 2/3 unused (2D)
    i32x8 z8 = {0, 0, 0, 0, 0, 0, 0, 0};
    __builtin_amdgcn_tensor_load_to_lds(g0, g1, z4, z4, z8, 0);
}

// ---------------------------------------------------------------------------
// 1) Row-normalize: one 128-thread block per row; gridDim.y selects array.
// ---------------------------------------------------------------------------
__global__ __launch_bounds__(128) void normalize_rows(const float* __restrict__ zi,
                                                      const float* __restrict__ zt,
                                                      float* __restrict__ An,
                                                      float* __restrict__ Bn) {
    __shared__ float red[128];
    const int row = blockIdx.x;
    const int t   = threadIdx.x;
    const float* src = blockIdx.y ? zt : zi;
    float*       dst = blockIdx.y ? Bn : An;

    const float4 v = *(const float4*)&src[row * DIM + t * 4];
    red[t] = v.x * v.x + v.y * v.y + v.z * v.z + v.w * v.w;
    __syncthreads();
    #pragma unroll
    for (int off = 64; off > 0; off >>= 1) {
        if (t < off) red[t] += red[t + off];
        __syncthreads();
    }
    const float scale = 1.0f / fmaxf(sqrtf(red[0]), EPS);
    float4 o;
    o.x = v.x * scale; o.y = v.y * scale; o.z = v.z * scale; o.w = v.w * scale;
    *(float4*)&dst[row * DIM + t * 4] = o;
}

// ---------------------------------------------------------------------------
// 2) Zero Srow|Scol accumulators
// ---------------------------------------------------------------------------
__global__ void zero_accum(float* __restrict__ p, int n) {
    int i = blockIdx.x * blockDim.x + threadIdx.x;
    if (i < n) p[i] = 0.0f;
}

// ---------------------------------------------------------------------------
// 3) Fused WMMA f32 GEMM + exp + row/col reduction + diagonal capture.
//    256 threads = 8 waves in a 4(M) x 2(N) grid; wave tile 32x64.
//    K double-buffered through LDS via TDM tensor_load_to_lds DMAs.
//    LDS layout: lA0 | lA1 | lB0 | lB1 | lrow(128) | lcol(128)
// ---------------------------------------------------------------------------
__global__ __launch_bounds__(256) void ntxent_gemm(const float* __restrict__ An,
                                                   const float* __restrict__ Bn,
                                                   float* __restrict__ Srow,
                                                   float* __restrict__ Scol,
                                                   float* __restrict__ Diag) {
    __shared__ float smem[4 * TILE_F + 256];
    float* const lrow = smem + 4 * TILE_F;
    float* const lcol = smem + 4 * TILE_F + BM;

    const int t       = threadIdx.x;
    const int rowBase = blockIdx.y * BM;
    const int colBase = blockIdx.x * BN;
    const int w     = t >> 5;
    const int wm    = w >> 1;        // 0..3 : 32-row strip
    const int wn    = w & 1;         // 0..1 : 64-col strip
    const int lane  = t & 31;
    const int laneM = lane & 15;
    const int hi    = lane >> 4;     // half-wave: K-pair / M+8 selector

    if (t < BM) lrow[t] = 0.0f; else lcol[t - BM] = 0.0f;

    v8f acc[2][4];
    #pragma unroll
    for (int mi = 0; mi < 2; ++mi)
        #pragma unroll
        for (int ni = 0; ni < 4; ++ni)
            acc[mi][ni] = (v8f){0.f,0.f,0.f,0.f,0.f,0.f,0.f,0.f};

    // prologue: DMA first K-block into buffer 0
    if (w == 0) {
        tdm_load_tile(0u, &An[(size_t)rowBase * DIM]);                       // lA0
        tdm_load_tile((unsigned)(2 * TILE_F * 4), &Bn[(size_t)colBase * DIM]); // lB0
        __builtin_amdgcn_s_wait_tensorcnt(0);
    }
    __syncthreads();

    for (int kb = 0; kb < DIM; kb += KB) {
        const int buf = (kb >> 5) & 1;
        const bool more = (kb + KB) < DIM;
        if (w == 0 && more) {   // overlap next K-block DMA with this compute
            const unsigned nb = (unsigned)(buf ^ 1);
            tdm_load_tile(nb * (unsigned)(TILE_F * 4),
                          &An[(size_t)rowBase * DIM + kb + KB]);
            tdm_load_tile((unsigned)(2 * TILE_F * 4) + nb * (unsigned)(TILE_F * 4),
                          &Bn[(size_t)colBase * DIM + kb + KB]);
        }

        const float* __restrict__ pa0 =
            smem + buf * TILE_F + (wm * 32 + laneM) * LDT + 2 * hi;
        const float* __restrict__ pa1 = pa0 + 16 * LDT;
        const float* __restrict__ pb0 =
            smem + (2 + buf) * TILE_F + (wn * 64 + laneM) * LDT + 2 * hi;

        #pragma unroll
        for (int k0 = 0; k0 < KB; k0 += 4) {
            // batch all fragment loads for this k-step, then burst 8 WMMAs
            v2f a[2], b[4];
            a[0].x = pa0[k0];            a[0].y = pa0[k0 + 1];
            a[1].x = pa1[k0];            a[1].y = pa1[k0 + 1];
            #pragma unroll
            for (int ni = 0; ni < 4; ++ni) {
                const float* pb = pb0 + ni * 16 * LDT;
                b[ni].x = pb[k0];        b[ni].y = pb[k0 + 1];
            }
            #pragma unroll
            for (int mi = 0; mi < 2; ++mi)
                #pragma unroll
                for (int ni = 0; ni < 4; ++ni)
                    acc[mi][ni] = __builtin_amdgcn_wmma_f32_16x16x4_f32(
                        false, a[mi], false, b[ni], (short)0, acc[mi][ni], false, false);
        }

        if (w == 0 && more) __builtin_amdgcn_s_wait_tensorcnt(0);
        __syncthreads();
    }

    // ---- Epilogue: e = exp(s/T); row/col sums; diagonal capture -----------
    float rsum[2][8];
    #pragma unroll
    for (int mi = 0; mi < 2; ++mi)
        #pragma unroll
        for (int r = 0; r < 8; ++r) rsum[mi][r] = 0.0f;

    #pragma unroll
    for (int ni = 0; ni < 4; ++ni) {
        const int gn = colBase + wn * 64 + ni * 16 + laneM;
        float csum = 0.0f;
        #pragma unroll
        for (int mi = 0; mi < 2; ++mi) {
            #pragma unroll
            for (int r = 0; r < 8; ++r) {
                const float s = acc[mi][ni][r] * INV_TEMP;
                const float e = __expf(s);
                rsum[mi][r] += e;
                csum        += e;
                const int gm = rowBase + wm * 32 + mi * 16 + r + hi * 8;
                if (gm == gn) Diag[gm] = s;   // unique writer
            }
        }
        // fold the two half-waves (same laneM) before the LDS atomic
        csum += __shfl_xor(csum, 16, 32);
        if (hi == 0) atomicAdd(&lcol[wn * 64 + ni * 16 + laneM], csum);
    }
    #pragma unroll
    for (int mi = 0; mi < 2; ++mi) {
        #pragma unroll
        for (int r = 0; r < 8; ++r) {
            float v = rsum[mi][r];   // butterfly across the 16 lanes of the half
            v += __shfl_xor(v, 1, 32);
            v += __shfl_xor(v, 2, 32);
            v += __shfl_xor(v, 4, 32);
            v += __shfl_xor(v, 8, 32);
            if (laneM == 0)
                atomicAdd(&lrow[wm * 32 + mi * 16 + r + hi * 8], v);
        }
    }

    __syncthreads();
    if (t < BM)  atomicAdd(&Srow[rowBase + t], lrow[t]);
    else         atomicAdd(&Scol[colBase + t - BM], lcol[t - BM]);
}

// ---------------------------------------------------------------------------
// 4) Final reduction: loss = (1/N) sum_i [ a*log(Srow) + (1-a)*log(Scol) - s_ii ]
// ---------------------------------------------------------------------------
__global__ __launch_bounds__(256) void finalize_loss(const float* __restrict__ Srow,
                                                     const float* __restrict__ Scol,
                                                     const float* __restrict__ Diag,
                                                     float* __restrict__ out) {
    __shared__ float red[256];
    const int t = threadIdx.x;
    float s = 0.0f;
    for (int i = t; i < BATCH; i += 256)
        s += ALPHA * __logf(Srow[i]) + (1.0f - ALPHA) * __logf(Scol[i]) - Diag[i];
    red[t] = s;
    __syncthreads();
    #pragma unroll
    for (int off = 128; off > 0; off >>= 1) {
        if (t < off) red[t] += red[t + off];
        __syncthreads();
    }
    if (t == 0) out[0] = red[0] / (float)BATCH;
}

// ---------------------------------------------------------------------------
extern "C" void kernel_launch(void* const* d_in, const int* in_sizes, int n_in,
                              void* d_out, int out_size, void* d_ws, size_t ws_size,
                              hipStream_t stream) {
    const float* z_img  = (const float*)d_in[0];
    const float* z_text = (const float*)d_in[1];
    float* out = (float*)d_out;

    float* ws   = (float*)d_ws;
    float* An   = ws;
    float* Bn   = An + (size_t)BATCH * DIM;
    float* Srow = Bn + (size_t)BATCH * DIM;
    float* Scol = Srow + BATCH;     // contiguous with Srow
    float* Diag = Scol + BATCH;

    dim3 gN(BATCH, 2);
    normalize_rows<<<gN, 128, 0, stream>>>(z_img, z_text, An, Bn);

    zero_accum<<<(2 * BATCH + 255) / 256, 256, 0, stream>>>(Srow, 2 * BATCH);

    dim3 gG(BATCH / BN, BATCH / BM);
    ntxent_gemm<<<gG, 256, 0, stream>>>(An, Bn, Srow, Scol, Diag);

    finalize_loss<<<1, 256, 0, stream>>>(Srow, Scol, Diag, out);
}